// HashEncoder_34230889349428
// MI455X (gfx1250) — compile-verified
//
#include <hip/hip_runtime.h>
#include <math.h>

#define NL 16
#define THREADS 256
#define PPT 8               // points per thread -> 512 workgroups for 1M points
#define L0_ENTRIES 4096     // level-0 table: 4096 float2 entries = 32 KB

typedef float vfloat2 __attribute__((ext_vector_type(2)));
typedef float vfloat4 __attribute__((ext_vector_type(4)));

// ---- Level constants, verified against the reference's float64 formula ----
// res_i = ceil(16*B^i - 1) + 1,  B = 1.3195079565048218 (slightly > 2^0.4, so
// levels 5/10/15 land just above 64/256/1024 with >=1e-5 relative margin).
__device__ constexpr unsigned RES[NL] = {
  16, 22, 28, 37, 49, 65, 85, 112, 148, 195, 257, 338, 446, 589, 777, 1025 };
// size = min(2^19, round8(res^3)); dense (under-hash) iff res^3 <= size.
__device__ constexpr unsigned SIZE[NL] = {
  4096, 10648, 21952, 50656, 117656, 274632,
  524288, 524288, 524288, 524288, 524288, 524288, 524288, 524288, 524288, 524288 };
__device__ constexpr unsigned OFF[NL] = {
  0, 4096, 14744, 36696, 87352, 205008, 479640, 1003928, 1528216,
  2052504, 2576792, 3101080, 3625368, 4149656, 4673944, 5198232 };
#define DENSE_MASK 0x003Fu   // levels 0..5 dense, 6..15 hashed (size = 2^19)
#define P1 2654435761u
#define P2 805459861u

struct Scales { float s[NL]; };   // only runtime-ish constants left (64 B kernarg)

// 32-bit LDS byte offset of a __shared__ object (addrspace(3) pointers are 32-bit).
__device__ __forceinline__ unsigned lds_byte_offset(const void* p) {
  return (unsigned)(unsigned long long)(__attribute__((address_space(3))) const void*)p;
}

// gfx1250 async global->LDS DMA (ASYNCcnt-tracked). Per lane: 16 bytes.
__device__ __forceinline__ void async_copy_b128_to_lds(unsigned lds_off, const void* gaddr) {
  asm volatile("global_load_async_to_lds_b128 %0, %1, off"
               :
               : "v"(lds_off), "v"(gaddr)
               : "memory");
}

__global__ __launch_bounds__(THREADS)
void hash_encode_kernel(const float* __restrict__ pos,
                        const float* __restrict__ table,
                        float* __restrict__ out,
                        unsigned n, Scales sc)
{
  __shared__ vfloat2 lds0[L0_ENTRIES];
  const vfloat2* __restrict__ tab2 = (const vfloat2*)table;

  const unsigned tid = threadIdx.x;

  // ---- Stage level-0 table (32 KB) into LDS with async global->LDS DMA ----
  {
    const unsigned base = lds_byte_offset(&lds0[0]);
    #pragma unroll
    for (int k = 0; k < (L0_ENTRIES * 8) / (THREADS * 16); ++k) {  // 8 sweeps
      unsigned chunk = (unsigned)k * THREADS + tid;                // 16B chunks
      async_copy_b128_to_lds(base + chunk * 16u,
                             (const char*)table + (size_t)chunk * 16u);
    }
    asm volatile("s_wait_asynccnt 0x0" ::: "memory");
  }
  __syncthreads();

  const unsigned blockBase = blockIdx.x * (THREADS * PPT);

  for (int pp = 0; pp < PPT; ++pp) {
    unsigned pidx = blockBase + (unsigned)pp * THREADS + tid;
    if (pidx >= n) continue;

    // Prefetch next iteration's positions (gfx1250 global_prefetch_b8; speculative-safe).
    __builtin_prefetch(pos + (size_t)(pidx + THREADS) * 3u, 0, 1);

    // Positions are streamed once: non-temporal so they don't evict the L2-resident table.
    float px = __builtin_nontemporal_load(pos + (size_t)pidx * 3u + 0);
    float py = __builtin_nontemporal_load(pos + (size_t)pidx * 3u + 1);
    float pz = __builtin_nontemporal_load(pos + (size_t)pidx * 3u + 2);

    // One fully-unrolled level: all per-level constants fold to immediates.
    auto do_level = [&](int l) -> vfloat2 {
      const float    s    = sc.s[l];
      const unsigned res  = RES[l];
      const unsigned size = SIZE[l];
      const unsigned off  = OFF[l];
      const bool     dense = (DENSE_MASK >> l) & 1u;   // compile-time per unrolled l

      // pos*scale + 0.5 with no FMA contraction (floor() boundary sensitivity)
      float xf = __fadd_rn(__fmul_rn(px, s), 0.5f);
      float yf = __fadd_rn(__fmul_rn(py, s), 0.5f);
      float zf = __fadd_rn(__fmul_rn(pz, s), 0.5f);
      float fx = floorf(xf), fy = floorf(yf), fz = floorf(zf);
      float tx = xf - fx,    ty = yf - fy,    tz = zf - fz;
      unsigned ix = (unsigned)fx, iy = (unsigned)fy, iz = (unsigned)fz;

      // per-axis partials hoisted out of the 8-corner loop
      const float wx[2] = { 1.0f - tx, tx };
      const float wy[2] = { 1.0f - ty, ty };
      const float wz[2] = { 1.0f - tz, tz };
      const float wxy[4] = { wx[0]*wy[0], wx[1]*wy[0], wx[0]*wy[1], wx[1]*wy[1] };

      unsigned ax[2], ay[2], az[2];
      if (dense) {
        ax[0] = ix;            ax[1] = ix + 1u;
        ay[0] = iy * res;      ay[1] = ay[0] + res;
        az[0] = iz * res*res;  az[1] = az[0] + res*res;
      } else {
        ax[0] = ix;            ax[1] = ix + 1u;
        ay[0] = iy * P1;       ay[1] = ay[0] + P1;
        az[0] = iz * P2;       az[1] = az[0] + P2;
      }

      unsigned ent[8];
      #pragma unroll
      for (int c = 0; c < 8; ++c) {
        const unsigned bx = c & 1u, by = (c >> 1) & 1u, bz = (c >> 2) & 1u;
        if (dense) {
          // under-index < 2*size  =>  "% size" == branchless min(u, u-size)
          unsigned u = ax[bx] + ay[by] + az[bz];
          ent[c] = min(u, u - size);
        } else {
          // hashed levels all have size == 2^19 (power of two)
          ent[c] = (ax[bx] ^ ay[by] ^ az[bz]) & (size - 1u);
        }
      }

      vfloat2 acc = {0.0f, 0.0f};
      vfloat2 f[8];
      if (l == 0) {
        #pragma unroll
        for (int c = 0; c < 8; ++c) f[c] = lds0[ent[c]];   // ds_load_b64 gathers
      } else {
        #pragma unroll
        for (int c = 0; c < 8; ++c) f[c] = tab2[off + ent[c]];  // batched b64 gathers
      }
      #pragma unroll
      for (int c = 0; c < 8; ++c)
        acc += f[c] * (wxy[c & 3u] * wz[(c >> 2) & 1u]);   // ((wx*wy)*wz), order-matched
      return acc;
    };

    // 8 non-temporal b128 stores per point (write-once; protects table in L2)
    vfloat4* __restrict__ out4 = (vfloat4*)out + (size_t)pidx * (NL / 2);
    #pragma unroll
    for (int h = 0; h < NL / 2; ++h) {
      vfloat2 a = do_level(2 * h);
      vfloat2 b = do_level(2 * h + 1);
      vfloat4 o = { a.x, a.y, b.x, b.y };
      __builtin_nontemporal_store(o, out4 + h);
    }
  }
}

extern "C" void kernel_launch(void* const* d_in, const int* in_sizes, int n_in,
                              void* d_out, int out_size, void* d_ws, size_t ws_size,
                              hipStream_t stream) {
  (void)n_in; (void)out_size; (void)d_ws; (void)ws_size;
  const float* pos   = (const float*)d_in[0];
  const float* table = (const float*)d_in[1];
  float*       out   = (float*)d_out;
  const unsigned n   = (unsigned)(in_sizes[0] / 3);

  // scale_l = 16 * exp_f32(l * f32(log(B))) - 1, matching the reference's f32 math
  Scales sc;
  const double lnB = log(1.3195079565048218);
  const float logBf = (float)lnB;
  for (int i = 0; i < NL; ++i) {
    float tf = (float)i * logBf;
    sc.s[i] = 16.0f * expf(tf) - 1.0f;
  }

  const unsigned per_block = THREADS * PPT;
  const unsigned blocks = (n + per_block - 1) / per_block;
  hash_encode_kernel<<<blocks, THREADS, 0, stream>>>(pos, table, out, n, sc);
}